// PatchSlicedWassersteinLoss_5145370820962
// MI455X (gfx1250) — compile-verified
//
#include <hip/hip_runtime.h>
#include <hip/hip_bf16.h>

// Problem constants (match reference)
#define BB   16
#define CC   4
#define HH   512
#define WW   512
#define PATCH 3
#define FEAT 36            // C * PATCH * PATCH
#define NPROJ 64
#define NS   4096          // MAX_SAMPLES
#define HW   (HH * WW)

typedef __attribute__((ext_vector_type(2))) float v2f;
typedef __attribute__((ext_vector_type(8))) float v8f;

// ---------------------------------------------------------------------------
// 1) Normalize theta columns: th[k,n] = theta[k,n] / ||theta[:,n]||
// ---------------------------------------------------------------------------
__global__ void k_normalize_theta(const float* __restrict__ theta,
                                  float* __restrict__ th) {
    int n = threadIdx.x;              // blockDim.x == 64
    if (n >= NPROJ) return;
    float ss = 0.f;
    for (int k = 0; k < FEAT; ++k) {
        float v = theta[k * NPROJ + n];
        ss += v * v;
    }
    float inv = rsqrtf(ss);
    for (int k = 0; k < FEAT; ++k)
        th[k * NPROJ + n] = theta[k * NPROJ + n] * inv;
}

// ---------------------------------------------------------------------------
// 2) Gather 3x3 zero-padded patches at sampled locations for both tensors.
//    Output: [NS, FEAT] row-major, feature order (c, ki, kj) like F.unfold.
// ---------------------------------------------------------------------------
__global__ void k_gather_patches(const float* __restrict__ xpred,
                                 const float* __restrict__ xstyle,
                                 const int*   __restrict__ indices,
                                 float* __restrict__ pp,
                                 float* __restrict__ ps) {
    int t = blockIdx.x * blockDim.x + threadIdx.x;
    if (t >= NS * FEAT) return;
    int s = t / FEAT;
    int f = t % FEAT;
    int c  = f / (PATCH * PATCH);
    int r  = f % (PATCH * PATCH);
    int ki = r / PATCH;
    int kj = r % PATCH;

    int idx = indices[s];
    int b   = idx / HW;
    int pos = idx % HW;
    int i   = pos / WW;
    int j   = pos % WW;

    int ii = i + ki - 1;
    int jj = j + kj - 1;

    float vp = 0.f, vs = 0.f;
    if (ii >= 0 && ii < HH && jj >= 0 && jj < WW) {
        size_t off = (((size_t)b * CC + c) * HH + ii) * WW + jj;
        vp = xpred[off];
        vs = xstyle[off];
    }
    pp[s * FEAT + f] = vp;
    ps[s * FEAT + f] = vs;
}

// ---------------------------------------------------------------------------
// 3) WMMA GEMM: [NS x 36] * [36 x 64] -> [NS x 64] for pred AND style.
//    One wave (32 threads) per 16x16 output tile; K looped in steps of 4
//    via V_WMMA_F32_16X16X4_F32. B (theta) loads are shared by both tensors.
//
//    f32 16x16x4 operand layout (ISA 7.12.2):
//      A (16x4): lanes 0-15 -> M=lane, VGPR0/1 = K0/K1;
//                lanes 16-31 -> M=lane-16, VGPR0/1 = K2/K3
//      B (4x16): mirrored with N over lanes
//      C/D f32 : VGPR r -> M=r (lanes 0-15) / M=r+8 (lanes 16-31), N=lane%16
// ---------------------------------------------------------------------------
__global__ void k_gemm_wmma(const float* __restrict__ Ap,   // [NS,36] pred patches
                            const float* __restrict__ As,   // [NS,36] style patches
                            const float* __restrict__ Bm,   // [36,64] normalized theta
                            float* __restrict__ Dp,         // [NS,64]
                            float* __restrict__ Ds) {       // [NS,64]
    const int tm   = blockIdx.x;          // 0..255  (M tiles)
    const int tn   = blockIdx.y;          // 0..3    (N tiles)
    const int lane = threadIdx.x;         // 0..31
    const int half = lane >> 4;           // 0: K=(0,1), 1: K=(2,3)
    const int l16  = lane & 15;

    const int m = tm * 16 + l16;          // A row for this lane
    const int n = tn * 16 + l16;          // B/D column for this lane

    v8f cp = {};
    v8f cs = {};

    #pragma unroll
    for (int k0 = 0; k0 < FEAT; k0 += 4) {
        const int ka = k0 + half * 2;     // first of this lane's two K values
        // A operands: float2 at 8B-aligned addresses (row stride 144B, ka even)
        v2f ap = *(const v2f*)(Ap + (size_t)m * FEAT + ka);
        v2f as = *(const v2f*)(As + (size_t)m * FEAT + ka);
        // B operand: two strided loads from theta (K-major rows of 64)
        v2f b;
        b.x = Bm[(ka + 0) * NPROJ + n];
        b.y = Bm[(ka + 1) * NPROJ + n];

        cp = __builtin_amdgcn_wmma_f32_16x16x4_f32(
                false, ap, false, b, (short)0, cp, false, false);
        cs = __builtin_amdgcn_wmma_f32_16x16x4_f32(
                false, as, false, b, (short)0, cs, false, false);
    }

    // Scatter the 16x16 f32 accumulators per the C/D VGPR layout.
    #pragma unroll
    for (int r = 0; r < 8; ++r) {
        const int mm = tm * 16 + r + half * 8;
        Dp[(size_t)mm * NPROJ + n] = cp[r];
        Ds[(size_t)mm * NPROJ + n] = cs[r];
    }
}

// ---------------------------------------------------------------------------
// 4) Per-column ascending bitonic sort of 4096 floats, fully LDS-resident.
//    grid.x = 128: blocks 0..63 sort pred columns, 64..127 style columns.
// ---------------------------------------------------------------------------
__global__ void k_bitonic_sort_col(float* __restrict__ proj_pred,
                                   float* __restrict__ proj_style) {
    __shared__ float s[NS];               // 16 KB
    float* buf = (blockIdx.x < NPROJ) ? proj_pred : proj_style;
    const int col = blockIdx.x & (NPROJ - 1);

    for (int i = threadIdx.x; i < NS; i += blockDim.x)
        s[i] = buf[(size_t)i * NPROJ + col];
    __syncthreads();

    for (int k = 2; k <= NS; k <<= 1) {
        for (int j = k >> 1; j > 0; j >>= 1) {
            for (int i = threadIdx.x; i < NS; i += blockDim.x) {
                int ixj = i ^ j;
                if (ixj > i) {
                    float a = s[i];
                    float b = s[ixj];
                    bool up = ((i & k) == 0);
                    if ((a > b) == up) { s[i] = b; s[ixj] = a; }
                }
            }
            __syncthreads();
        }
    }

    for (int i = threadIdx.x; i < NS; i += blockDim.x)
        buf[(size_t)i * NPROJ + col] = s[i];
}

// ---------------------------------------------------------------------------
// 5) Deterministic MSE reduction: single block, LDS tree (no float atomics).
// ---------------------------------------------------------------------------
__global__ void k_mse_reduce(const float* __restrict__ p,
                             const float* __restrict__ q,
                             float* __restrict__ out) {
    __shared__ float red[1024];
    float acc = 0.f;
    for (int i = threadIdx.x; i < NS * NPROJ; i += 1024) {
        float d = p[i] - q[i];
        acc += d * d;
    }
    red[threadIdx.x] = acc;
    __syncthreads();
    for (int s = 512; s > 0; s >>= 1) {
        if ((int)threadIdx.x < s) red[threadIdx.x] += red[threadIdx.x + s];
        __syncthreads();
    }
    if (threadIdx.x == 0)
        out[0] = red[0] / (float)(NS * NPROJ);
}

// ---------------------------------------------------------------------------
// Launch
// ---------------------------------------------------------------------------
extern "C" void kernel_launch(void* const* d_in, const int* in_sizes, int n_in,
                              void* d_out, int out_size, void* d_ws, size_t ws_size,
                              hipStream_t stream) {
    const float* x_pred  = (const float*)d_in[0];
    const float* x_style = (const float*)d_in[1];
    const float* theta   = (const float*)d_in[2];
    const int*   indices = (const int*)d_in[3];
    float* out = (float*)d_out;

    char* ws = (char*)d_ws;
    // Offsets (all 256B-aligned)
    float* pp     = (float*)(ws + 0);                       // [NS,36]  589824 B
    float* ps     = (float*)(ws + 589824);                  // [NS,36]  589824 B
    float* th     = (float*)(ws + 1179648);                 // [36,64]    9216 B
    float* proj_p = (float*)(ws + 1188864);                 // [NS,64] 1048576 B
    float* proj_s = (float*)(ws + 2237440);                 // [NS,64] 1048576 B
    // total: 3,286,016 B

    k_normalize_theta<<<1, 64, 0, stream>>>(theta, th);

    {
        int total = NS * FEAT;
        int blk = 256;
        k_gather_patches<<<(total + blk - 1) / blk, blk, 0, stream>>>(
            x_pred, x_style, indices, pp, ps);
    }

    {
        dim3 grid(NS / 16, NPROJ / 16);   // 256 x 4 tiles
        k_gemm_wmma<<<grid, 32, 0, stream>>>(pp, ps, th, proj_p, proj_s);
    }

    k_bitonic_sort_col<<<2 * NPROJ, 256, 0, stream>>>(proj_p, proj_s);

    k_mse_reduce<<<1, 1024, 0, stream>>>(proj_p, proj_s, out);
}